// Liu_U_61933428408753
// MI455X (gfx1250) — compile-verified
//
#include <hip/hip_runtime.h>

// CDNA5 / gfx1250, wave32. One wave per block; each wave owns a 16-row batch
// tile and runs the T=1024 recurrence with V_WMMA_F32_16X16X4_F32 (K=4
// matches the 4-wide u/x/hidden sub-blocks exactly; full fp32 precision).
//
// Software pipelined: measurement MLP of step t and dynamics MLP of step t+1
// both depend only on x_t, so they are issued together in one branch-free
// basic block (separate LDS transpose buffers) and interleave.
//
// Layouts (CDNA5 ISA):
//  A 16x4 f32 : lane = M row (lane&15); VGPR0 = K(2*(lane>>4)), VGPR1 = K+1
//  B 4x16 f32 : lane = N col (lane&15); VGPR0 = K(2*(lane>>4)), VGPR1 = K+1
//  C/D 16x16  : VGPR v: lanes 0-15 -> M=v, N=lane; lanes 16-31 -> M=v+8
//
// LDS row stride 36 floats: unconditional tile-b stores (cols 16..31) fit,
// and banking is conflict-free (36c mod 64 distinct; +8 rows ≡ +32 banks).

typedef float v2f __attribute__((ext_vector_type(2)));
typedef float v8f __attribute__((ext_vector_type(8)));

#define T_LEN 1024
#define HS 36          // LDS row stride (floats)
#define BUF (16 * HS)  // one transpose buffer

__device__ __forceinline__ v8f wmma4(v2f a, v2f b, v8f c) {
  return __builtin_amdgcn_wmma_f32_16x16x4_f32(false, a, false, b, (short)0, c,
                                               false, false);
}
// Single-instruction ReLU: v_max_num_f32 is IEEE maximumNumber (NaN-aware),
// no canonicalize needed. src0 = inline constant 0, src1 = VGPR (VOP2 rule).
__device__ __forceinline__ float relu1(float x) {
  float r;
  asm("v_max_num_f32 %0, 0, %1" : "=v"(r) : "v"(x));
  return r;
}

__global__ __launch_bounds__(32) void liu_rnn_kernel(
    const float* __restrict__ u, const float* __restrict__ y,
    const float* __restrict__ W1, const float* __restrict__ b1,
    const float* __restrict__ W2, const float* __restrict__ b2,
    const float* __restrict__ W3, const float* __restrict__ b3,
    const float* __restrict__ W4, const float* __restrict__ b4,
    float* __restrict__ partial) {
  __shared__ float sbuf[4 * BUF];

  const int lane = threadIdx.x & 31;
  const int col  = lane & 15;  // A: M row / B,C: N col
  const int hlf  = lane >> 4;
  const int b0   = blockIdx.x * 16;

  // ---- loop-invariant weight tiles in B layout ----
  auto mkB = [&](const float* W, int K, int N, int s, int nt) -> v2f {
    int k = 4 * s + 2 * hlf;
    int n = nt * 16 + col;
    v2f r;
    r.x = (n < N && k < K)     ? W[k * N + n]       : 0.0f;
    r.y = (n < N && k + 1 < K) ? W[(k + 1) * N + n] : 0.0f;
    return r;
  };
  v2f w1t[2][2], w3t[2][2], w2t[5], w4t[5];
  for (int s = 0; s < 2; ++s)
    for (int nt = 0; nt < 2; ++nt) {
      w1t[s][nt] = mkB(W1, 8, 20, s, nt);
      w3t[s][nt] = mkB(W3, 8, 20, s, nt);
    }
  for (int s = 0; s < 5; ++s) {
    w2t[s] = mkB(W2, 20, 4, s, 0);
    w4t[s] = mkB(W4, 20, 4, s, 0);
  }
  // bias broadcast into C-accumulator initializers
  float bv1a = b1[col];
  float bv1b = (col < 4) ? b1[16 + col] : 0.0f;
  float bv3a = b3[col];
  float bv3b = (col < 4) ? b3[16 + col] : 0.0f;
  float bv2  = (col < 4) ? b2[col] : 0.0f;
  float bv4  = (col < 4) ? b4[col] : 0.0f;
  v8f bi1a, bi1b, bi3a, bi3b, bi2, bi4;
#pragma unroll
  for (int i = 0; i < 8; ++i) {
    bi1a[i] = bv1a; bi1b[i] = bv1b;
    bi3a[i] = bv3a; bi3b[i] = bv3b;
    bi2[i]  = bv2;  bi4[i]  = bv4;
  }

  // per-lane global streams: u[b0+col, 2*hlf + {0,1}, t] (t-contiguous)
  const float* up = u + (size_t)(b0 + col) * (4 * T_LEN) + (size_t)(2 * hlf) * T_LEN;
  const float* yp = y + (size_t)(b0 + col) * (4 * T_LEN) + (size_t)(2 * hlf) * T_LEN;

  // hoisted per-buffer LDS base pointers (DS ops become base + immediate)
  const int cst = (8 * hlf) * HS + col;  // C store base; tile b at +16
  const int ald = col * HS + 2 * hlf;    // A load base; K-step s at +4s
  float* st_dh = sbuf + cst;             // dyn h1 store base
  float* st_dx = sbuf + BUF + cst;       // dyn x store base
  float* st_mh = sbuf + 2 * BUF + cst;   // meas h2 store base
  float* st_my = sbuf + 3 * BUF + cst;   // meas y_hat store base
  const float* ld_dh = sbuf + ald;
  const float* ld_dx = sbuf + BUF + ald;
  const float* ld_mh = sbuf + 2 * BUF + ald;
  const float* ld_my = sbuf + 3 * BUF + ald;

  v2f a_x = {0.0f, 0.0f};        // x_{t-1} tile (A layout)
  v2f a_u = {up[0], up[T_LEN]};  // u_t tile (A layout)

  // ---- prologue: dynamics(0) -> x_0 ----
  {
    v8f c1a = wmma4(a_u, w1t[0][0], bi1a);
    c1a = wmma4(a_x, w1t[1][0], c1a);
    v8f c1b = wmma4(a_u, w1t[0][1], bi1b);
    c1b = wmma4(a_x, w1t[1][1], c1b);
#pragma unroll
    for (int i = 0; i < 8; ++i) {
      st_dh[i * HS]      = relu1(c1a[i]);
      st_dh[16 + i * HS] = relu1(c1b[i]);
    }
    __builtin_amdgcn_wave_barrier();
    v8f c2 = bi2;
#pragma unroll
    for (int s = 0; s < 5; ++s)
      c2 = wmma4(*(const v2f*)(ld_dh + 4 * s), w2t[s], c2);
#pragma unroll
    for (int i = 0; i < 8; ++i) st_dx[i * HS] = c2[i];
    __builtin_amdgcn_wave_barrier();
    a_x = *(const v2f*)ld_dx;
  }

  float acc = 0.0f;

#pragma unroll 1
  for (int t = 0; t < T_LEN; ++t) {
    const int tn = (t + 1 < T_LEN) ? (t + 1) : (T_LEN - 1);  // uniform
    v2f a_un = {up[tn], up[tn + T_LEN]};                     // u_{t+1}
    float yv0 = yp[t], yv1 = yp[t + T_LEN];
    __builtin_prefetch(&up[t + 96], 0, 3);
    __builtin_prefetch(&up[t + 96 + T_LEN], 0, 3);
    __builtin_prefetch(&yp[t + 96], 0, 3);
    __builtin_prefetch(&yp[t + 96 + T_LEN], 0, 3);

    // ---- stage 1: meas(t) layer-3 and dyn(t+1) layer-1 (independent) ----
    v8f c3a = wmma4(a_u, w3t[0][0], bi3a);
    c3a = wmma4(a_x, w3t[1][0], c3a);
    v8f c3b = wmma4(a_u, w3t[0][1], bi3b);
    c3b = wmma4(a_x, w3t[1][1], c3b);
    v8f c1a = wmma4(a_un, w1t[0][0], bi1a);
    c1a = wmma4(a_x, w1t[1][0], c1a);
    v8f c1b = wmma4(a_un, w1t[0][1], bi1b);
    c1b = wmma4(a_x, w1t[1][1], c1b);

    // ---- stage 2: relu + transpose through LDS (branch-free) ----
#pragma unroll
    for (int i = 0; i < 8; ++i) {
      st_mh[i * HS]      = relu1(c3a[i]);
      st_mh[16 + i * HS] = relu1(c3b[i]);
      st_dh[i * HS]      = relu1(c1a[i]);
      st_dh[16 + i * HS] = relu1(c1b[i]);
    }
    __builtin_amdgcn_wave_barrier();
    v2f a4[5], a2[5];
#pragma unroll
    for (int s = 0; s < 5; ++s) {
      a4[s] = *(const v2f*)(ld_mh + 4 * s);
      a2[s] = *(const v2f*)(ld_dh + 4 * s);
    }

    // ---- stage 3: meas(t) layer-4 and dyn(t+1) layer-2 (independent) ----
    v8f c4 = bi4, c2 = bi2;
#pragma unroll
    for (int s = 0; s < 5; ++s) {
      c4 = wmma4(a4[s], w4t[s], c4);
      c2 = wmma4(a2[s], w2t[s], c2);
    }
#pragma unroll
    for (int i = 0; i < 8; ++i) {
      st_my[i * HS] = c4[i];
      st_dx[i * HS] = c2[i];
    }
    __builtin_amdgcn_wave_barrier();
    v2f yh = *(const v2f*)ld_my;  // y_hat[b=col][2*hlf + {0,1}]
    a_x = *(const v2f*)ld_dx;     // x_{t+1} feed-forward

    float d0 = yh.x - yv0, d1 = yh.y - yv1;
    acc = fmaf(d0, d0, acc);
    acc = fmaf(d1, d1, acc);
    a_u = a_un;
  }

  // deterministic fixed-order butterfly reduction over the wave
  for (int off = 16; off > 0; off >>= 1) acc += __shfl_xor(acc, off, 32);
  if (lane == 0) partial[blockIdx.x] = acc;
}

__global__ __launch_bounds__(32) void liu_reduce_kernel(
    const float* __restrict__ partial, float* __restrict__ out, int n) {
  float s = 0.0f;
  for (int i = threadIdx.x; i < n; i += 32) s += partial[i];
  for (int off = 16; off > 0; off >>= 1) s += __shfl_xor(s, off, 32);
  if (threadIdx.x == 0) out[0] = s;
}

extern "C" void kernel_launch(void* const* d_in, const int* in_sizes, int n_in,
                              void* d_out, int out_size, void* d_ws,
                              size_t ws_size, hipStream_t stream) {
  (void)in_sizes; (void)n_in; (void)out_size; (void)ws_size;
  const float* u  = (const float*)d_in[0];
  const float* y  = (const float*)d_in[1];
  const float* W1 = (const float*)d_in[2];
  const float* b1 = (const float*)d_in[3];
  const float* W2 = (const float*)d_in[4];
  const float* b2 = (const float*)d_in[5];
  const float* W3 = (const float*)d_in[6];
  const float* b3 = (const float*)d_in[7];
  const float* W4 = (const float*)d_in[8];
  const float* b4 = (const float*)d_in[9];
  float* partial = (float*)d_ws;  // 128 floats of scratch

  liu_rnn_kernel<<<128, 32, 0, stream>>>(u, y, W1, b1, W2, b2, W3, b3, W4, b4,
                                         partial);
  liu_reduce_kernel<<<1, 32, 0, stream>>>(partial, (float*)d_out, 128);
}